// GraphCNN_69621419868297
// MI455X (gfx1250) — compile-verified
//
#include <hip/hip_runtime.h>

// GIN forward for MI455X (gfx1250, wave32).
// Sparse aggregation + V_WMMA_F32_16X16X4_F32 GEMMs + BN/ReLU + readout.

#define NN   4096   // nodes
#define NE   65536  // edges
#define FE   4      // edge features
#define DIN  64
#define DH   128
#define DOUT 16
#define NG   32

typedef __attribute__((ext_vector_type(2))) float v2f;
typedef __attribute__((ext_vector_type(8))) float v8f;

// ---------------- utility ----------------
__global__ void k_zero(float* __restrict__ p, int n) {
    int i = blockIdx.x * blockDim.x + threadIdx.x;
    if (i < n) p[i] = 0.0f;
}

// diag[i][f] = 1.0 (identity self-loops)
__global__ void k_init_diag(float* __restrict__ diag) {
    int i = blockIdx.x * blockDim.x + threadIdx.x;
    if (i < NN) {
        float4 one = {1.0f, 1.0f, 1.0f, 1.0f};
        *(float4*)&diag[i * 4] = one;
    }
}

// self-loop edges overwrite the diagonal entry (reference .set semantics)
__global__ void k_selfloop(const int* __restrict__ ei, const float* __restrict__ attr,
                           float* __restrict__ diag) {
    int e = blockIdx.x * blockDim.x + threadIdx.x;
    if (e >= NE) return;
    int s = ei[e], d = ei[NE + e];
    if (s == d) *(float4*)&diag[s * 4] = *(const float4*)&attr[e * 4];
}

// pooled[i, d*4+f] = (diag[i,f] + eps[l]) * h[i,d]   (initializes pooled fully)
__global__ void k_pooled_diag(const float* __restrict__ h, const float* __restrict__ diag,
                              const float* __restrict__ epsv, int l, int D,
                              float* __restrict__ pooled) {
    int idx = blockIdx.x * blockDim.x + threadIdx.x;
    if (idx >= NN * D) return;
    int i = idx / D;
    float hv = h[idx];
    float e  = epsv[l];
    float4 dg = *(const float4*)&diag[i * 4];
    float4 o;
    o.x = (dg.x + e) * hv; o.y = (dg.y + e) * hv;
    o.z = (dg.z + e) * hv; o.w = (dg.w + e) * hv;
    *(float4*)&pooled[(size_t)idx * 4] = o;   // idx*4 == (i*D + d)*4
}

// pooled[src, d*4+f] += attr[e,f] * h[dst,d]  (sparse einsum; skip self-loops,
// they were folded into the diagonal term)
__global__ void k_pooled_edges(const int* __restrict__ ei, const float* __restrict__ attr,
                               const float* __restrict__ h, int D,
                               float* __restrict__ pooled) {
    int idx = blockIdx.x * blockDim.x + threadIdx.x;
    if (idx >= NE * D) return;
    int e = idx / D, d = idx - e * D;
    int s = ei[e], t = ei[NE + e];
    if (s == t) return;
    float hv = h[(size_t)t * D + d];
    float4 at = *(const float4*)&attr[e * 4];
    float* base = &pooled[(size_t)(s * D + d) * 4];
    atomicAdd(base + 0, at.x * hv);
    atomicAdd(base + 1, at.y * hv);
    atomicAdd(base + 2, at.z * hv);
    atomicAdd(base + 3, at.w * hv);
}

// ---------------- WMMA GEMM: out[M,Nout] = A[M,K] @ W[Nout,K]^T + bias ----------------
// Nout == 128: 8 waves per block, one 16x16 tile per wave; grid.x = M/16.
// Fragment layouts per CDNA5 ISA 7.12.2:
//   A 16x4 : m = lane&15, (k0,k1) = base + 2*(lane>>4) + {0,1}   -> one float2
//   B 4x16 : n = lane&15, (k0,k1) = base + 2*(lane>>4) + {0,1}   -> one float2 of W^T
//   C/D    : col n = lane&15, row = i + 8*(lane>>4), i = VGPR index 0..7
__global__ __launch_bounds__(256) void k_gemm_wmma_f32(
    const float* __restrict__ A, const float* __restrict__ W,
    const float* __restrict__ bias, float* __restrict__ out,
    int M, int K, int Nout) {
    int lane  = threadIdx.x & 31;
    int wave  = threadIdx.x >> 5;
    int half  = lane >> 4;     // 0: lanes 0-15, 1: lanes 16-31
    int lid   = lane & 15;
    int mbase = blockIdx.x * 16;
    int nbase = wave * 16;
    int m = mbase + lid;
    int n = nbase + lid;

    const float* arow = A + (size_t)m * K + 2 * half;
    const float* wrow = W + (size_t)n * K + 2 * half;

    v8f c = {0.f, 0.f, 0.f, 0.f, 0.f, 0.f, 0.f, 0.f};
    for (int k = 0; k < K; k += 4) {
        v2f a = *(const v2f*)(arow + k);
        v2f b = *(const v2f*)(wrow + k);
        // (neg_a, A, neg_b, B, c_mod, C, reuse_a, reuse_b)
        c = __builtin_amdgcn_wmma_f32_16x16x4_f32(false, a, false, b,
                                                  (short)0, c, false, false);
    }
    float bv = bias[n];
    int mrow = mbase + 8 * half;
    for (int i = 0; i < 8; ++i)
        out[(size_t)(mrow + i) * Nout + n] = c[i] + bv;
}

// ---------------- BatchNorm (training-mode, biased variance) ----------------
__global__ void k_bn_stats(const float* __restrict__ t, float* __restrict__ mean,
                           float* __restrict__ invstd) {
    int c = blockIdx.x;               // one block per channel, gridDim.x = DH
    __shared__ float ssum[256];
    __shared__ float ssq[256];
    float s = 0.f, q = 0.f;
    for (int i = threadIdx.x; i < NN; i += blockDim.x) {
        float v = t[(size_t)i * DH + c];
        s += v; q += v * v;
    }
    ssum[threadIdx.x] = s; ssq[threadIdx.x] = q;
    __syncthreads();
    for (int off = 128; off > 0; off >>= 1) {
        if ((int)threadIdx.x < off) {
            ssum[threadIdx.x] += ssum[threadIdx.x + off];
            ssq[threadIdx.x]  += ssq[threadIdx.x + off];
        }
        __syncthreads();
    }
    if (threadIdx.x == 0) {
        float mval = ssum[0] / (float)NN;
        float var  = ssq[0] / (float)NN - mval * mval;
        mean[c]   = mval;
        invstd[c] = rsqrtf(var + 1e-5f);
    }
}

__global__ void k_bn_relu(const float* __restrict__ in, const float* __restrict__ g,
                          const float* __restrict__ b, const float* __restrict__ mean,
                          const float* __restrict__ invstd, float* __restrict__ out) {
    int idx = blockIdx.x * blockDim.x + threadIdx.x;
    if (idx >= NN * DH) return;
    int c = idx & (DH - 1);
    float v = g[c] * (in[idx] - mean[c]) * invstd[c] + b[c];
    out[idx] = fmaxf(v, 0.0f);
}

// ---------------- readout ----------------
__global__ void k_segsum(const float* __restrict__ h, const int* __restrict__ batch,
                         float* __restrict__ acc, int D) {
    int idx = blockIdx.x * blockDim.x + threadIdx.x;
    if (idx >= NN * D) return;
    int i = idx / D, d = idx - i * D;
    atomicAdd(&acc[batch[i] * D + d], h[idx]);
}

__global__ void k_pred(const float* __restrict__ a0, const float* __restrict__ a1,
                       const float* __restrict__ a2,
                       const float* __restrict__ w0, const float* __restrict__ b0,
                       const float* __restrict__ w1, const float* __restrict__ b1,
                       const float* __restrict__ w2, const float* __restrict__ b2,
                       float* __restrict__ out) {
    int t = blockIdx.x * blockDim.x + threadIdx.x;
    if (t >= NG * DOUT) return;
    int g = t / DOUT, o = t - g * DOUT;
    float s = b0[o] + b1[o] + b2[o];
    for (int d = 0; d < DIN; ++d) s += a0[g * DIN + d] * w0[o * DIN + d];
    for (int d = 0; d < DH;  ++d) s += a1[g * DH  + d] * w1[o * DH  + d];
    for (int d = 0; d < DH;  ++d) s += a2[g * DH  + d] * w2[o * DH  + d];
    out[t] = s;
}

static inline int cdiv(long n, int b) { return (int)((n + b - 1) / b); }

extern "C" void kernel_launch(void* const* d_in, const int* in_sizes, int n_in,
                              void* d_out, int out_size, void* d_ws, size_t ws_size,
                              hipStream_t stream) {
    // inputs (setup_inputs dict order, pytrees flattened depth-first in insertion order)
    const float* x      = (const float*)d_in[0];
    const int*   ei     = (const int*)  d_in[1];   // [2,E] row-major: src then dst
    const float* attr   = (const float*)d_in[2];
    const int*   batch  = (const int*)  d_in[3];
    const float* epsv   = (const float*)d_in[4];   // [2]
    const float* m0_l0w = (const float*)d_in[5];   // [128,256]
    const float* m0_l0b = (const float*)d_in[6];
    const float* m0_g   = (const float*)d_in[7];
    const float* m0_b   = (const float*)d_in[8];
    const float* m0_l1w = (const float*)d_in[9];   // [128,128]
    const float* m0_l1b = (const float*)d_in[10];
    const float* m1_l0w = (const float*)d_in[11];  // [128,512]
    const float* m1_l0b = (const float*)d_in[12];
    const float* m1_g   = (const float*)d_in[13];
    const float* m1_b   = (const float*)d_in[14];
    const float* m1_l1w = (const float*)d_in[15];  // [128,128]
    const float* m1_l1b = (const float*)d_in[16];
    const float* bn0g   = (const float*)d_in[17];
    const float* bn0b   = (const float*)d_in[18];
    const float* bn1g   = (const float*)d_in[19];
    const float* bn1b   = (const float*)d_in[20];
    const float* p0w    = (const float*)d_in[21];  // [16,64]
    const float* p0b    = (const float*)d_in[22];
    const float* p1w    = (const float*)d_in[23];  // [16,128]
    const float* p1b    = (const float*)d_in[24];
    const float* p2w    = (const float*)d_in[25];  // [16,128]
    const float* p2b    = (const float*)d_in[26];
    float* out = (float*)d_out;

    // workspace carve-up (~17 MB of f32)
    float* ws     = (float*)d_ws;
    float* diag   = ws;                       // NN*4           = 16384
    float* pooled = diag   + NN * 4;          // NN*512         = 2097152
    float* t0     = pooled + NN * 512;        // NN*128
    float* t1     = t0     + NN * DH;         // NN*128
    float* h1     = t1     + NN * DH;         // NN*128
    float* h2     = h1     + NN * DH;         // NN*128
    float* mean   = h2     + NN * DH;         // 128
    float* invstd = mean   + DH;              // 128
    float* acc0   = invstd + DH;              // 32*64
    float* acc1   = acc0   + NG * DIN;        // 32*128
    float* acc2   = acc1   + NG * DH;         // 32*128

    const int B = 256;

    // adjacency diagonal (shared by both layers; eps added per layer)
    k_init_diag<<<cdiv(NN, B), B, 0, stream>>>(diag);
    k_selfloop <<<cdiv(NE, B), B, 0, stream>>>(ei, attr, diag);

    // ----- layer 0: h_in = x (D=64) -----
    k_pooled_diag <<<cdiv((long)NN * DIN, B), B, 0, stream>>>(x, diag, epsv, 0, DIN, pooled);
    k_pooled_edges<<<cdiv((long)NE * DIN, B), B, 0, stream>>>(ei, attr, x, DIN, pooled);
    k_gemm_wmma_f32<<<NN / 16, 256, 0, stream>>>(pooled, m0_l0w, m0_l0b, t0, NN, DIN * FE, DH);
    k_bn_stats<<<DH, 256, 0, stream>>>(t0, mean, invstd);
    k_bn_relu <<<cdiv((long)NN * DH, B), B, 0, stream>>>(t0, m0_g, m0_b, mean, invstd, t0);
    k_gemm_wmma_f32<<<NN / 16, 256, 0, stream>>>(t0, m0_l1w, m0_l1b, t1, NN, DH, DH);
    k_bn_stats<<<DH, 256, 0, stream>>>(t1, mean, invstd);
    k_bn_relu <<<cdiv((long)NN * DH, B), B, 0, stream>>>(t1, bn0g, bn0b, mean, invstd, h1);

    // ----- layer 1: h_in = h1 (D=128) -----
    k_pooled_diag <<<cdiv((long)NN * DH, B), B, 0, stream>>>(h1, diag, epsv, 1, DH, pooled);
    k_pooled_edges<<<cdiv((long)NE * DH, B), B, 0, stream>>>(ei, attr, h1, DH, pooled);
    k_gemm_wmma_f32<<<NN / 16, 256, 0, stream>>>(pooled, m1_l0w, m1_l0b, t0, NN, DH * FE, DH);
    k_bn_stats<<<DH, 256, 0, stream>>>(t0, mean, invstd);
    k_bn_relu <<<cdiv((long)NN * DH, B), B, 0, stream>>>(t0, m1_g, m1_b, mean, invstd, t0);
    k_gemm_wmma_f32<<<NN / 16, 256, 0, stream>>>(t0, m1_l1w, m1_l1b, t1, NN, DH, DH);
    k_bn_stats<<<DH, 256, 0, stream>>>(t1, mean, invstd);
    k_bn_relu <<<cdiv((long)NN * DH, B), B, 0, stream>>>(t1, bn1g, bn1b, mean, invstd, h2);

    // ----- readout -----
    int accTotal = NG * DIN + 2 * NG * DH;
    k_zero<<<cdiv(accTotal, B), B, 0, stream>>>(acc0, accTotal);  // acc0..acc2 contiguous
    k_segsum<<<cdiv((long)NN * DIN, B), B, 0, stream>>>(x,  batch, acc0, DIN);
    k_segsum<<<cdiv((long)NN * DH,  B), B, 0, stream>>>(h1, batch, acc1, DH);
    k_segsum<<<cdiv((long)NN * DH,  B), B, 0, stream>>>(h2, batch, acc2, DH);
    k_pred<<<cdiv(NG * DOUT, B), B, 0, stream>>>(acc0, acc1, acc2,
                                                 p0w, p0b, p1w, p1b, p2w, p2b, out);
}